// My_GNN_NN_37915971289093
// MI455X (gfx1250) — compile-verified
//
#include <hip/hip_runtime.h>
#include <hip/hip_bf16.h>
#include <cstdint>

#define N_NODES  50000
#define FEAT_IN  64
#define FEAT_OUT 32
#define HIDDEN   128
#define N_EDGES  800000
#define T_LEN    (N_NODES * FEAT_OUT)   // 1,600,000
#define OUT_LEN  (2 * N_NODES)          // 100,000
#define M_TILES  (N_NODES / 16)         // 3125 (exact)

typedef float v2f __attribute__((ext_vector_type(2)));
typedef float v4f __attribute__((ext_vector_type(4)));
typedef float v8f __attribute__((ext_vector_type(8)));

// ---------- 1. degree / norm ----------
__global__ void init_deg_kernel(float* __restrict__ deg) {
    int i = blockIdx.x * blockDim.x + threadIdx.x;
    if (i < N_NODES) deg[i] = 1.0f;            // self-loop
}

__global__ void scatter_deg_kernel(const long long* __restrict__ ei,
                                   float* __restrict__ deg) {
    int e = blockIdx.x * blockDim.x + threadIdx.x;
    if (e < N_EDGES) {
        int d = (int)ei[N_EDGES + e];          // dst row
        atomicAdd(&deg[d], 1.0f);
    }
}

__global__ void rsqrt_deg_kernel(float* __restrict__ deg) {
    int i = blockIdx.x * blockDim.x + threadIdx.x;
    if (i < N_NODES) deg[i] = __frsqrt_rn(deg[i]);   // deg>0 always (self-loop)
}

// ---------- 2. h = x @ conv_w via V_WMMA_F32_16X16X4_F32 ----------
// One wave -> one 16-row tile, 32 output cols (two 16x16 accumulators).
__global__ __launch_bounds__(256) void gcn_xw_wmma_kernel(
        const float* __restrict__ x, const float* __restrict__ w,
        float* __restrict__ h) {
    const int lane = threadIdx.x & 31;
    const int wave = threadIdx.x >> 5;
    const int tile = blockIdx.x * 8 + wave;
    if (tile >= M_TILES) return;
    const int half = lane >> 4;     // 0: K pair {0,1}, 1: K pair {2,3}
    const int m    = lane & 15;

#if __has_builtin(__builtin_amdgcn_wmma_f32_16x16x4_f32)
    const int row = tile * 16 + m;
    v8f c0 = {}; v8f c1 = {};
#pragma unroll
    for (int kk = 0; kk < FEAT_IN; kk += 4) {
        const int k = kk + 2 * half;
        // A 16x4: lane (half,m) holds x[row][k], x[row][k+1]  (contiguous pair)
        v2f a;
        a.x = x[row * FEAT_IN + k];
        a.y = x[row * FEAT_IN + k + 1];
        // B 4x16: lane n holds w[k][n], w[k+1][n]
        v2f b0, b1;
        b0.x = w[k * FEAT_OUT + m];
        b0.y = w[(k + 1) * FEAT_OUT + m];
        b1.x = w[k * FEAT_OUT + 16 + m];
        b1.y = w[(k + 1) * FEAT_OUT + 16 + m];
        c0 = __builtin_amdgcn_wmma_f32_16x16x4_f32(false, a, false, b0,
                                                   (short)0, c0, false, false);
        c1 = __builtin_amdgcn_wmma_f32_16x16x4_f32(false, a, false, b1,
                                                   (short)0, c1, false, false);
    }
    // C/D layout: lanes 0-15 -> M=v, N=lane; lanes 16-31 -> M=v+8, N=lane-16
#pragma unroll
    for (int v = 0; v < 8; ++v) {
        const int r = tile * 16 + v + 8 * half;
        h[r * FEAT_OUT + m]      = c0[v];
        h[r * FEAT_OUT + 16 + m] = c1[v];
    }
#else
    // Scalar fallback (keeps compile alive if builtin is absent)
    const int row = tile * 16 + m;
#pragma unroll 4
    for (int cc = 0; cc < 16; ++cc) {
        const int col = half * 16 + cc;
        float acc = 0.f;
        for (int k = 0; k < FEAT_IN; ++k)
            acc = fmaf(x[row * FEAT_IN + k], w[k * FEAT_OUT + col], acc);
        h[row * FEAT_OUT + col] = acc;
    }
#endif
}

// ---------- 3. aggregation ----------
// Self-loop contribution: agg = h * dinv^2
__global__ void init_agg_kernel(const float* __restrict__ h,
                                const float* __restrict__ dinv,
                                float* __restrict__ agg) {
    int i = blockIdx.x * blockDim.x + threadIdx.x;
    if (i < T_LEN) {
        float d = dinv[i >> 5];
        agg[i] = h[i] * d * d;
    }
}

// One lane per (edge, feature): 8 edges per 256-thread block
__global__ __launch_bounds__(256) void edge_scatter_kernel(
        const long long* __restrict__ ei, const float* __restrict__ h,
        const float* __restrict__ dinv, float* __restrict__ agg) {
    const int e = blockIdx.x * 8 + (threadIdx.x >> 5);
    const int f = threadIdx.x & 31;
    if (e < N_EDGES) {
        const int s = (int)ei[e];
        const int d = (int)ei[N_EDGES + e];
        const float nrm = dinv[s] * dinv[d];
        atomicAdd(&agg[d * FEAT_OUT + f], h[s * FEAT_OUT + f] * nrm);
    }
}

// t = tanh(agg + conv_b), in place
__global__ void tanh_agg_kernel(float* __restrict__ agg,
                                const float* __restrict__ cb) {
    int i = blockIdx.x * blockDim.x + threadIdx.x;
    if (i < T_LEN) agg[i] = tanhf(agg[i] + cb[i & 31]);
}

// ---------- 4. y = tanh(lin1_w @ t + lin1_b) ----------
__global__ void zero_y_kernel(float* __restrict__ y_raw) {
    int i = blockIdx.x * blockDim.x + threadIdx.x;
    if (i < HIDDEN) y_raw[i] = 0.0f;
}

// grid (chunks, HIDDEN); each block reduces a 4096-elt slice of one row.
// lin1_w (819 MB) is streamed once with non-temporal 128-bit loads.
__global__ __launch_bounds__(256) void lin1_partial_kernel(
        const float* __restrict__ w, const float* __restrict__ t,
        float* __restrict__ y_raw) {
    const int row   = blockIdx.y;
    const int tid   = threadIdx.x;
    const size_t ro = (size_t)row * T_LEN;
    const int base  = blockIdx.x * 4096 + tid * 4;
    float acc = 0.f;
#pragma unroll
    for (int it = 0; it < 4; ++it) {
        const int i = base + it * 1024;
        if (i < T_LEN) {
            v4f wv = __builtin_nontemporal_load((const v4f*)(w + ro + i));
            v4f tv = *(const v4f*)(t + i);
            acc += wv.x * tv.x + wv.y * tv.y + wv.z * tv.z + wv.w * tv.w;
        }
    }
    __shared__ float red[256];
    red[tid] = acc;
    __syncthreads();
#pragma unroll
    for (int s = 128; s > 0; s >>= 1) {
        if (tid < s) red[tid] += red[tid + s];
        __syncthreads();
    }
    if (tid == 0) atomicAdd(&y_raw[row], red[0]);
}

__global__ void y_act_kernel(const float* __restrict__ y_raw,
                             const float* __restrict__ b,
                             float* __restrict__ y) {
    int i = blockIdx.x * blockDim.x + threadIdx.x;
    if (i < HIDDEN) y[i] = tanhf(y_raw[i] + b[i]);
}

// ---------- 5. out = lin2_w @ y + lin2_b ----------
__global__ __launch_bounds__(256) void lin2_kernel(
        const float* __restrict__ w, const float* __restrict__ y,
        const float* __restrict__ b, float* __restrict__ out) {
    __shared__ float ys[HIDDEN];
    if (threadIdx.x < HIDDEN) ys[threadIdx.x] = y[threadIdx.x];
    __syncthreads();
    const int o = blockIdx.x * 256 + threadIdx.x;
    if (o >= OUT_LEN) return;
    const v4f* wr = (const v4f*)(w + (size_t)o * HIDDEN);
    float acc = b[o];
#pragma unroll
    for (int k = 0; k < HIDDEN / 4; ++k) {
        v4f wv = wr[k];
        acc += wv.x * ys[4 * k] + wv.y * ys[4 * k + 1] +
               wv.z * ys[4 * k + 2] + wv.w * ys[4 * k + 3];
    }
    out[o] = acc;
}

extern "C" void kernel_launch(void* const* d_in, const int* in_sizes, int n_in,
                              void* d_out, int out_size, void* d_ws, size_t ws_size,
                              hipStream_t stream) {
    const float*     x      = (const float*)d_in[0];
    const long long* ei     = (const long long*)d_in[1];   // int64 [2, E]
    const float*     conv_w = (const float*)d_in[2];
    const float*     conv_b = (const float*)d_in[3];
    const float*     lin1_w = (const float*)d_in[4];
    const float*     lin1_b = (const float*)d_in[5];
    const float*     lin2_w = (const float*)d_in[6];
    const float*     lin2_b = (const float*)d_in[7];
    float*           out    = (float*)d_out;

    // workspace layout (floats): deg/dinv | h | agg(t) | y_raw | y
    float* deg   = (float*)d_ws;
    float* h     = deg + N_NODES;       // 16B-aligned (200000 bytes)
    float* agg   = h + T_LEN;
    float* y_raw = agg + T_LEN;
    float* y     = y_raw + HIDDEN;

    // 1. normalization coefficients
    init_deg_kernel<<<(N_NODES + 255) / 256, 256, 0, stream>>>(deg);
    scatter_deg_kernel<<<(N_EDGES + 255) / 256, 256, 0, stream>>>(ei, deg);
    rsqrt_deg_kernel<<<(N_NODES + 255) / 256, 256, 0, stream>>>(deg);

    // 2. h = x @ conv_w  (WMMA f32, 8 tiles per block)
    gcn_xw_wmma_kernel<<<(M_TILES + 7) / 8, 256, 0, stream>>>(x, conv_w, h);

    // 3. message passing
    init_agg_kernel<<<(T_LEN + 255) / 256, 256, 0, stream>>>(h, deg, agg);
    edge_scatter_kernel<<<(N_EDGES + 7) / 8, 256, 0, stream>>>(ei, h, deg, agg);
    tanh_agg_kernel<<<(T_LEN + 255) / 256, 256, 0, stream>>>(agg, conv_b);

    // 4. hidden layer (split-K reduction over the 819 MB weight stream)
    zero_y_kernel<<<1, HIDDEN, 0, stream>>>(y_raw);
    dim3 g1((T_LEN + 4095) / 4096, HIDDEN);
    lin1_partial_kernel<<<g1, 256, 0, stream>>>(lin1_w, agg, y_raw);
    y_act_kernel<<<1, HIDDEN, 0, stream>>>(y_raw, lin1_b, y);

    // 5. output layer
    lin2_kernel<<<(OUT_LEN + 255) / 256, 256, 0, stream>>>(lin2_w, y, lin2_b, out);
}